// ControlFlowScanDecomposition_151564_46308337386065
// MI455X (gfx1250) — compile-verified
//
#include <hip/hip_runtime.h>

// out[i, j] = images[i, j] if j < pos[i] else 0
// Bandwidth-bound prefix-mask copy for MI455X:
//  - 128-bit coalesced NT loads/stores (global_load_b128 / global_store_b128, TH=NT;
//    output (256MB) exceeds the 192MB L2, so stream both directions)
//  - wave-uniform scalar load of pos[row] (s_load_b64)
//  - load-skipping for fully-masked vec4 chunks (wave-coherent branch; ~halves
//    average HBM read traffic since pos ~ U[1, L-1])
//  - 32-bit compares (p < L fits in i32), 2 vec4s per thread to amortize
//    addressing overhead and double loads-in-flight per wave.

typedef long long i64;
// Native clang vector type: accepted by __builtin_nontemporal_* and lowers to
// b128 global memory ops. 16-byte aligned like HIP float4.
typedef __attribute__((ext_vector_type(4))) float v4f;

#define VEC_PER_THREAD 2

__global__ __launch_bounds__(256)
void prefix_mask_copy_kernel(const v4f* __restrict__ img,
                             const i64* __restrict__ pos,
                             v4f*       __restrict__ out,
                             int col4_per_row) {
    const int row = blockIdx.y;                         // one row per grid.y
    // blockIdx.y is workgroup-uniform -> uniform address -> s_load_b64.
    // 0 < pos < L <= 2^31, so truncate to i32 and keep all compares 32-bit.
    const int p = (int)pos[row];

    const size_t rowbase = (size_t)row * (size_t)col4_per_row;
    const int    base    = blockIdx.x * (256 * VEC_PER_THREAD) + threadIdx.x;

#pragma unroll
    for (int k = 0; k < VEC_PER_THREAD; ++k) {
        const int col4 = base + k * 256;                // coalesced per wave
        if (col4 >= col4_per_row) continue;

        const int    j   = col4 << 2;                   // starting column
        const size_t idx = rowbase + (size_t)col4;

        v4f v;
        if (j >= p) {
            // Entire vec4 past the prefix: zeros WITHOUT loading. A wave spans
            // 128 consecutive columns of one row, so this branch is
            // wave-coherent except for the single boundary wave per row.
            v = (v4f)(0.f);
        } else {
            v = __builtin_nontemporal_load(&img[idx]);  // global_load_b128, NT
            if (j + 4 > p) {
                // Boundary vector (at most one per row): mask tail lanes.
                // j < p here, so v.x is always kept.
                if (j + 1 >= p) v.y = 0.f;
                if (j + 2 >= p) v.z = 0.f;
                if (j + 3 >= p) v.w = 0.f;
            }
        }
        __builtin_nontemporal_store(v, &out[idx]);      // global_store_b128, NT
    }
}

extern "C" void kernel_launch(void* const* d_in, const int* in_sizes, int n_in,
                              void* d_out, int out_size, void* d_ws, size_t ws_size,
                              hipStream_t stream) {
    const float* img = (const float*)d_in[0];   // [N, L] fp32
    const i64*   pos = (const i64*)d_in[1];     // [N]    int64 (per reference dtype)
    float*       out = (float*)d_out;           // [N, L] fp32

    const int N    = in_sizes[1];               // 16384
    const int L    = in_sizes[0] / N;           // 4096 (divisible by 4)
    const int col4 = L >> 2;                    // 1024 vec4 per row

    const int col4_per_block = 256 * VEC_PER_THREAD;    // 512
    dim3 block(256);                                    // 8 wave32 waves
    dim3 grid((col4 + col4_per_block - 1) / col4_per_block, N);  // (2, 16384)

    prefix_mask_copy_kernel<<<grid, block, 0, stream>>>(
        (const v4f*)img, pos, (v4f*)out, col4);
}